// SpatialEncoder_16578573762771
// MI455X (gfx1250) — compile-verified
//
#include <hip/hip_runtime.h>

// ---------------------------------------------------------------------------
// Event-camera spatial histogram: 20M events [x,y,t,pol] -> (2,24,32) density.
// Bandwidth-bound (320 MB stream / 23.3 TB/s ~= 14 us). No matrix math ->
// WMMA not applicable; optimize the wave32 / LDS / L2-atomic path instead.
// 8x-unrolled grid-stride stream: ~33 MB of B128 loads in flight grid-wide,
// comfortably above the ~16 MB (BW x latency) needed to saturate HBM.
// ---------------------------------------------------------------------------

#define BINS_X    32
#define BINS_Y    24
#define BINS_POL  2
#define BINS_TOT  (BINS_POL * BINS_Y * BINS_X)   // 1536
#define NCOPIES   8                              // one LDS histogram per wave32
#define BLOCK     256                            // 8 waves of 32
#define GRID_HIST 1024
#define UNROLL    8

__device__ __forceinline__ void bump_event(unsigned int* __restrict__ myHist,
                                           float4 e) {
    // x_factor = 32/640, y_factor = 24/480 (f32 multiply, truncate like .astype(int32))
    int xb = (int)(e.x * (32.0f / 640.0f));
    int yb = (int)(e.y * (24.0f / 480.0f));
    xb = min(max(xb, 0), BINS_X - 1);
    yb = min(max(yb, 0), BINS_Y - 1);
    int pol  = (e.w > 0.0f) ? 0 : 1;             // channel 0 = pos, 1 = neg
    int flat = pol * (BINS_Y * BINS_X) + yb * BINS_X + xb;
    atomicAdd(&myHist[flat], 1u);                // ds_add_u32 (no return)
}

__global__ __launch_bounds__(BLOCK) void zero_ws_kernel(unsigned int* __restrict__ ws) {
    int i = blockIdx.x * BLOCK + threadIdx.x;
    if (i < BINS_TOT) ws[i] = 0u;
}

__global__ __launch_bounds__(BLOCK)
void hist_kernel(const float4* __restrict__ ev,
                 unsigned int* __restrict__ ghist,
                 int n) {
    __shared__ unsigned int lhist[NCOPIES * BINS_TOT];   // 48 KB of 320 KB WGP LDS

    // Zero the privatized LDS histograms.
    for (int i = threadIdx.x; i < NCOPIES * BINS_TOT; i += BLOCK) lhist[i] = 0u;
    __syncthreads();

    const unsigned wave = threadIdx.x >> 5;              // wave32 on gfx1250
    unsigned int* __restrict__ my = lhist + wave * BINS_TOT;

    const int stride = (int)gridDim.x * BLOCK;
    int i = (int)blockIdx.x * BLOCK + (int)threadIdx.x;

    // Main loop: UNROLL outstanding global_load_b128 per thread; one
    // global_prefetch_b8 (counter-free L2 hint) one batch ahead.
    for (; i + (UNROLL - 1) * stride < n; i += UNROLL * stride) {
        __builtin_prefetch(&ev[i + UNROLL * stride], 0, 0);
        float4 e[UNROLL];
#pragma unroll
        for (int k = 0; k < UNROLL; ++k) e[k] = ev[i + k * stride];
#pragma unroll
        for (int k = 0; k < UNROLL; ++k) bump_event(my, e[k]);
    }
    for (; i < n; i += stride) {
        bump_event(my, ev[i]);
    }
    __syncthreads();

    // One flush per block: merge the 8 wave-private copies, single L2 atomic
    // per bin per block (1024 * 1536 atomics total across the grid; ~1k
    // serialized per address -> sub-microsecond).
    for (int b = threadIdx.x; b < BINS_TOT; b += BLOCK) {
        unsigned int s = 0;
#pragma unroll
        for (int c = 0; c < NCOPIES; ++c) s += lhist[c * BINS_TOT + b];
        if (s) atomicAdd(&ghist[b], s);                  // global_atomic_add_u32
    }
}

__global__ __launch_bounds__(BLOCK)
void finalize_kernel(const unsigned int* __restrict__ ghist,
                     float* __restrict__ out) {
    __shared__ unsigned int red[BLOCK];

    // 1536 bins / 256 threads = 6 bins each; exact integer total.
    unsigned int v[BINS_TOT / BLOCK];
    unsigned int s = 0;
#pragma unroll
    for (int k = 0; k < BINS_TOT / BLOCK; ++k) {
        v[k] = ghist[threadIdx.x + k * BLOCK];
        s += v[k];
    }
    red[threadIdx.x] = s;
    __syncthreads();
    for (int off = BLOCK / 2; off > 0; off >>= 1) {
        if ((int)threadIdx.x < off) red[threadIdx.x] += red[threadIdx.x + off];
        __syncthreads();
    }
    float total = (float)red[0];                         // 20,000,000: exact in f32
#pragma unroll
    for (int k = 0; k < BINS_TOT / BLOCK; ++k) {
        float h = (float)v[k];
        out[threadIdx.x + k * BLOCK] = (total > 0.0f) ? (h / total) : h;
    }
}

extern "C" void kernel_launch(void* const* d_in, const int* in_sizes, int n_in,
                              void* d_out, int out_size, void* d_ws, size_t ws_size,
                              hipStream_t stream) {
    (void)n_in; (void)out_size; (void)ws_size;

    const float4* events = (const float4*)d_in[0];
    const int n_events   = in_sizes[0] / 4;              // [N,4] f32, flat count / 4
    unsigned int* counts = (unsigned int*)d_ws;          // 1536 u32 scratch
    float* out           = (float*)d_out;                // 2*24*32 f32

    // 1) Zero the accumulator (harness poisons d_ws; must be deterministic).
    zero_ws_kernel<<<(BINS_TOT + BLOCK - 1) / BLOCK, BLOCK, 0, stream>>>(counts);

    // 2) Privatized LDS histogram + block-level L2 atomic flush.
    hist_kernel<<<GRID_HIST, BLOCK, 0, stream>>>(events, counts, n_events);

    // 3) Exact total + normalize into d_out.
    finalize_kernel<<<1, BLOCK, 0, stream>>>(counts, out);
}